// FourierLayer_64596308132492
// MI455X (gfx1250) — compile-verified
//
#include <hip/hip_runtime.h>

typedef __attribute__((ext_vector_type(2))) float v2f;
typedef __attribute__((ext_vector_type(4))) float v4f;
typedef __attribute__((ext_vector_type(8))) float v8f;

#define SIZE 1024
#define MTOT (16 * 1024)
#define BM 64
#define BN 64
#define BK 64
#define LDA 68   // padded LDS row stride (floats): 272B = 8B-aligned, stride-4 banks

#define PI_D 3.14159265358979323846264338327950288

// --------------------------------------------------------------------------
// Feature gate: Tensor Data Mover path (gfx1250 TDM), with safe fallback.
// --------------------------------------------------------------------------
#if defined(__has_builtin)
#if __has_builtin(__builtin_amdgcn_tensor_load_to_lds) && \
    __has_builtin(__builtin_amdgcn_s_wait_tensorcnt)
#define FL_USE_TDM 1
#endif
#endif

#ifdef FL_USE_TDM
#define NBUF 2
#else
#define NBUF 1
#endif

// ---------------------------------------------------------------------------
// Kernel 1: sr = sum(weights[...,0]), si = sum(weights[...,1])
// ---------------------------------------------------------------------------
__global__ void __launch_bounds__(256)
fl_reduce_weights(const float* __restrict__ w, float* __restrict__ sums) {
    __shared__ float s0[256];
    __shared__ float s1[256];
    const int t = threadIdx.x;
    float sr = 0.f, si = 0.f;
    for (int i = t; i < 64 * 64; i += 256) {
        sr += w[2 * i + 0];
        si += w[2 * i + 1];
    }
    s0[t] = sr; s1[t] = si;
    __syncthreads();
    for (int off = 128; off > 0; off >>= 1) {
        if (t < off) { s0[t] += s0[t + off]; s1[t] += s1[t + off]; }
        __syncthreads();
    }
    if (t == 0) { sums[0] = s0[0]; sums[1] = s1[0]; }
}

// ---------------------------------------------------------------------------
// Kernel 2: W[p,m] = si * H[(m-p) mod 1024],
//           H[d] = -(2/N) * cot(pi*d/N) for odd d, else 0.
// (Exact closed form of irfft(i * rfft(e_p))[m], including the C2R
//  treatment of bins 0 and N/2.)
// ---------------------------------------------------------------------------
__global__ void __launch_bounds__(256)
fl_build_w(const float* __restrict__ sums, float* __restrict__ W) {
    const float si = sums[1];
    const int p = blockIdx.x;
    for (int m = threadIdx.x; m < SIZE; m += blockDim.x) {
        const int d = (m - p) & (SIZE - 1);
        float h = 0.f;
        if (d & 1) {
            const double a = PI_D * (double)d / (double)SIZE;
            h = (float)(-(2.0 / (double)SIZE) / tan(a));
        }
        W[(size_t)p * SIZE + m] = si * h;
    }
}

// ---------------------------------------------------------------------------
// TDM helper: issue TENSOR_LOAD_TO_LDS of a 2-D tile (64 x 64 f32) with
// hardware LDS padding of 4 DWORDs after every 64 DWORDs (-> [64][68] rows).
// D# packing per CDNA5 ISA 8.3/8.4; groups 2/3 zero (2-D tensor).
// ---------------------------------------------------------------------------
#ifdef FL_USE_TDM
typedef __attribute__((ext_vector_type(4))) unsigned int u32x4;
typedef __attribute__((ext_vector_type(4))) int          i32x4;
typedef __attribute__((ext_vector_type(8))) int          i32x8;

__device__ __forceinline__ void
fl_tdm_load_tile64(const float* gtile, unsigned lds_addr,
                   unsigned tensor_d0, unsigned tensor_d1, unsigned stride0) {
    const unsigned long long ga = (unsigned long long)(size_t)gtile;

    u32x4 g0;
    g0.x = 1u;                                          // count=1, user desc
    g0.y = lds_addr;                                    // [63:32] lds_addr
    g0.z = (unsigned)(ga & 0xFFFFFFFFu);                // global_addr[31:0]
    g0.w = (unsigned)((ga >> 32) & 0x1FFFFFFu)          // global_addr[56:32]
         | (2u << 30);                                  // type = 2 ("image")

    i32x8 g1;
    g1[0] = (int)((2u << 16)                            // data_size = 4B
                | (1u << 20)                            // pad_enable
                | (5u << 22)                            // pad_interval: 64 DW
                | (3u << 25));                          // pad_amount:   4 DW
    g1[1] = (int)((tensor_d0 & 0xFFFFu) << 16);         // [47:32]=0, td0 lo16
    g1[2] = (int)(((tensor_d0 >> 16) & 0xFFFFu) | ((tensor_d1 & 0xFFFFu) << 16));
    g1[3] = (int)(((tensor_d1 >> 16) & 0xFFFFu) | (64u << 16));   // tile_dim0=64
    g1[4] = (int)(64u);                                 // tile_dim1=64, tile_dim2=0
    g1[5] = (int)stride0;                               // dim0 stride [31:0]
    g1[6] = 0;                                          // stride0 hi, stride1 lo = 0
    g1[7] = 0;                                          // stride1 hi = 0

    const i32x4 z4 = {};
#if __clang_major__ >= 23
    const i32x8 z8 = {};
    __builtin_amdgcn_tensor_load_to_lds(g0, g1, z4, z4, z8, 0);
#else
    __builtin_amdgcn_tensor_load_to_lds(g0, g1, z4, z4, 0);
#endif
}
#endif  // FL_USE_TDM

// ---------------------------------------------------------------------------
// Kernel 3: out = sr * x + x @ W   via V_WMMA_F32_16X16X4_F32
//   grid: (N/BN, MTOT/BM), 256 threads = 8 wave32
//   wave w: N-tile tn = w&3 (16 cols), M-half tm2 = w>>2 (32 rows as 2 tiles)
//   TDM path: waves 0/1 DMA the A/B tiles, double-buffered, s_wait_tensorcnt.
// ---------------------------------------------------------------------------
__global__ void __launch_bounds__(256)
fl_hilbert_gemm(const float* __restrict__ X, const float* __restrict__ W,
                const float* __restrict__ sums, float* __restrict__ out) {
    __shared__ float As[NBUF][BM][LDA];
    __shared__ float Bs[NBUF][BK][LDA];

    const int tid   = threadIdx.x;
    const int lane  = tid & 31;
    const int wave  = tid >> 5;
    const int m0    = blockIdx.y * BM;
    const int n0    = blockIdx.x * BN;

    const int tn    = wave & 3;            // which 16-wide N tile
    const int tm2   = wave >> 2;           // which 32-row M half
    const int lmod  = lane & 15;
    const int khalf = (lane >> 4) << 1;    // 0 or 2  (A/B K split per lane half)
    const int mhalf = (lane >> 4) << 3;    // 0 or 8  (C/D M split per lane half)

    v8f acc0 = {};
    v8f acc1 = {};

#ifdef FL_USE_TDM
    const unsigned ldsA0    = (unsigned)(unsigned long long)(size_t)&As[0][0][0];
    const unsigned ldsB0    = (unsigned)(unsigned long long)(size_t)&Bs[0][0][0];
    const unsigned bufBytes = (unsigned)(BM * LDA * sizeof(float));
    // Prologue: DMA chunk 0 into buffer 0.
    if (wave == 0)
        fl_tdm_load_tile64(X + (size_t)m0 * SIZE, ldsA0, SIZE, MTOT, SIZE);
    else if (wave == 1)
        fl_tdm_load_tile64(W + n0, ldsB0, SIZE, SIZE, SIZE);
#endif

    for (int k0 = 0; k0 < SIZE; k0 += BK) {
        const int cur = (k0 / BK) & (NBUF - 1);

#ifdef FL_USE_TDM
        const int k1 = k0 + BK;
        if (k1 < SIZE) {
            const unsigned nb = (unsigned)(cur ^ 1) * bufBytes;
            if (wave == 0)
                fl_tdm_load_tile64(X + (size_t)m0 * SIZE + k1, ldsA0 + nb,
                                   SIZE, MTOT, SIZE);
            else if (wave == 1)
                fl_tdm_load_tile64(W + (size_t)k1 * SIZE + n0, ldsB0 + nb,
                                   SIZE, SIZE, SIZE);
            if (wave < 2) __builtin_amdgcn_s_wait_tensorcnt(1);  // cur done
        } else {
            if (wave < 2) __builtin_amdgcn_s_wait_tensorcnt(0);
        }
        __syncthreads();
#else
        // Fallback: cooperative global->LDS staging, float4 per slot
        #pragma unroll
        for (int q = 0; q < 4; ++q) {
            const int idx  = tid + q * 256;      // 0..1023 float4 slots
            const int row  = idx >> 4;
            const int col4 = (idx & 15) << 2;
            const v4f av = *(const v4f*)(X + (size_t)(m0 + row) * SIZE + k0 + col4);
            As[0][row][col4 + 0] = av.x; As[0][row][col4 + 1] = av.y;
            As[0][row][col4 + 2] = av.z; As[0][row][col4 + 3] = av.w;
            const v4f bv = *(const v4f*)(W + (size_t)(k0 + row) * SIZE + n0 + col4);
            Bs[0][row][col4 + 0] = bv.x; Bs[0][row][col4 + 1] = bv.y;
            Bs[0][row][col4 + 2] = bv.z; Bs[0][row][col4 + 3] = bv.w;
        }
        __syncthreads();
#endif

        #pragma unroll
        for (int kk = 0; kk < BK; kk += 4) {
            // B 4x16 frag: v0 = row (kk+khalf), v1 = row (kk+khalf+1), N = lmod
            v2f bfrag;
            bfrag.x = Bs[cur][kk + khalf + 0][tn * 16 + lmod];
            bfrag.y = Bs[cur][kk + khalf + 1][tn * 16 + lmod];
            // A 16x4 frags: lane M = lmod, K pair (khalf, khalf+1), 8B-aligned
            const float* ap0 = &As[cur][tm2 * 32 +      lmod][kk + khalf];
            const float* ap1 = &As[cur][tm2 * 32 + 16 + lmod][kk + khalf];
            v2f a0; a0.x = ap0[0]; a0.y = ap0[1];
            v2f a1; a1.x = ap1[0]; a1.y = ap1[1];
            acc0 = __builtin_amdgcn_wmma_f32_16x16x4_f32(
                false, a0, false, bfrag, (short)0, acc0, false, false);
            acc1 = __builtin_amdgcn_wmma_f32_16x16x4_f32(
                false, a1, false, bfrag, (short)0, acc1, false, false);
        }
        __syncthreads();
    }

    // Epilogue: out = sr*x + acc   (C/D layout: vgpr e -> M = e + mhalf)
    const float sr = sums[0];
    const int nG = n0 + tn * 16 + lmod;
    #pragma unroll
    for (int e = 0; e < 8; ++e) {
        const int mG0 = m0 + tm2 * 32 +      e + mhalf;
        const int mG1 = m0 + tm2 * 32 + 16 + e + mhalf;
        const size_t i0 = (size_t)mG0 * SIZE + nG;
        const size_t i1 = (size_t)mG1 * SIZE + nG;
        out[i0] = sr * X[i0] + acc0[e];
        out[i1] = sr * X[i1] + acc1[e];
    }
}

// ---------------------------------------------------------------------------
extern "C" void kernel_launch(void* const* d_in, const int* in_sizes, int n_in,
                              void* d_out, int out_size, void* d_ws, size_t ws_size,
                              hipStream_t stream) {
    (void)in_sizes; (void)n_in; (void)out_size; (void)ws_size;
    const float* x = (const float*)d_in[0];   // (16, 1024, 1024) f32
    const float* w = (const float*)d_in[1];   // (64, 64, 2) f32
    float* out  = (float*)d_out;              // (16, 1024, 1024) f32
    float* sums = (float*)d_ws;               // [0]=sr, [1]=si
    float* W    = sums + 64;                  // 256B offset; 1024x1024 f32 (4MB)

    fl_reduce_weights<<<1, 256, 0, stream>>>(w, sums);
    fl_build_w<<<SIZE, 256, 0, stream>>>(sums, W);
    dim3 grid(SIZE / BN, MTOT / BM);
    fl_hilbert_gemm<<<grid, 256, 0, stream>>>(x, W, sums, out);
}